// BahdanauAttention_27882927686396
// MI455X (gfx1250) — compile-verified
//
#include <hip/hip_runtime.h>
#include <math.h>

// CDNA5 (gfx1250) Bahdanau attention: one workgroup per batch element.
// - Window gather + W1 panel staging via Tensor Data Mover (tensor_load_to_lds)
// - Core GEMM [128,512]x[512,512] per batch on V_WMMA_F32_16X16X4_F32
// - tanh/V/softmax/context fused in the same kernel.

typedef __attribute__((ext_vector_type(2))) float v2f;
typedef __attribute__((ext_vector_type(8))) float v8f;
typedef __attribute__((ext_vector_type(4))) unsigned int u32x4;
typedef __attribute__((ext_vector_type(4))) int i32x4;
typedef __attribute__((ext_vector_type(8))) int i32x8;

#define B_    64
#define T_    4096
#define U_    512
#define L_    128
#define HALF_ 64

// Window tile in LDS: TDM pads 2 DWORDs after every 256 DWORDs
// (pad_interval=7, pad_amount=1) -> addr(l,u) = l*516 + u + 2*(u>=256).
// 516 mod 64 = 4 banks => conflict-free 16-lane b64 fragment reads.
#define ATT_STRIDE 516
#define ATT_FLOATS (L_ * ATT_STRIDE)   // 66048
#define PAN_FLOATS (U_ * 16)           // 8192, panel stored [k][16n], no pad

// Issue a 2D TDM tile load: global (td0 x td1, row stride stride0 elems,
// 4-byte elements) -> LDS at lds_byte. OOB rows/cols read as zero.
__device__ __forceinline__ void tdm_load_2d(unsigned lds_byte,
                                            unsigned long long gaddr,
                                            unsigned td0, unsigned td1,
                                            unsigned tile0, unsigned tile1,
                                            unsigned stride0,
                                            unsigned dw0_flags) {
  u32x4 g0 = {
      1u,                                     // count=1, user descriptor
      lds_byte,                               // lds_addr
      (unsigned)gaddr,                        // global_addr[31:0]
      (unsigned)(gaddr >> 32) | (2u << 30)    // global_addr[56:32] | type=2
  };
  i32x8 g1 = {
      (int)dw0_flags,                         // data_size=4B | pad ctl
      (int)(td0 << 16),                       // abar_addr=0 | tensor_dim0 lo
      (int)((td0 >> 16) | (td1 << 16)),       // tensor_dim0 hi | tensor_dim1 lo
      (int)((td1 >> 16) | (tile0 << 16)),     // tensor_dim1 hi | tile_dim0
      (int)(tile1 & 0xffffu),                 // tile_dim1 | tile_dim2=0
      (int)stride0,                           // tensor_dim0_stride lo32
      0,                                      // stride0 hi | dim1_stride lo
      0
  };
  i32x4 z4 = {0, 0, 0, 0};
  i32x8 z8 = {0, 0, 0, 0, 0, 0, 0, 0};
  // 6-arg form (this toolchain): (g0, g1, g2, g3, g4, cpol)
  __builtin_amdgcn_tensor_load_to_lds(g0, g1, z4, z4, z8, 0);
}

__global__ __launch_bounds__(256) void bahdanau_wmma_kernel(
    const float* __restrict__ query,   // [B,U]
    const float* __restrict__ values,  // [B,T,U]
    const int*   __restrict__ pos,     // [B]
    const float* __restrict__ W1,      // [U,U]
    const float* __restrict__ b1,      // [U]
    const float* __restrict__ W2,      // [U,U]
    const float* __restrict__ b2,      // [U]
    const float* __restrict__ V,       // [U,1]
    float* __restrict__ out)           // [B*U] context ++ [B*L] weights
{
  extern __shared__ float smem[];
  float* att   = smem;                 // [128][516] padded window (TDM)
  float* panel = att + ATT_FLOATS;     // [512][16]  W1 panel (TDM)
  float* qsh   = panel + PAN_FLOATS;   // [512]
  float* q2    = qsh + U_;             // [512]
  float* score = q2 + U_;              // [128]
  float* wts   = score + L_;           // [128]
  float* red   = wts + L_;             // [2]

  const int tid  = threadIdx.x;
  const int b    = blockIdx.x;
  const int wave = tid >> 5;
  const int lane = tid & 31;
  const int lmod = lane & 15;          // A: row-in-strip / B,C: column
  const int kh   = lane >> 4;          // K-pair select
  const int mrow = 16 * wave + lmod;

  const unsigned ldsbase = __builtin_amdgcn_groupstaticsize();

  const int p     = pos[b];
  const int start = max(p - HALF_, 0);
  const int end   = min(p + HALF_, T_);
  const int xlen  = end - start;       // padding always at the tail

  // ---- stage query row ----
  qsh[tid]       = query[b * U_ + tid];
  qsh[tid + 256] = query[b * U_ + tid + 256];

  // ---- TDM: gather window -> LDS; rows >= xlen read as zero (OOB),
  //      LDS padding inserted by TDM (2 dw / 256 dw). Wave 0 issues. ----
  if (wave == 0) {
    unsigned long long ga =
        (unsigned long long)(uintptr_t)(values + ((size_t)b * T_ + start) * U_);
    unsigned galo = (unsigned)__builtin_amdgcn_readfirstlane((int)ga);
    unsigned gahi = (unsigned)__builtin_amdgcn_readfirstlane((int)(ga >> 32));
    unsigned xl   = (unsigned)__builtin_amdgcn_readfirstlane(xlen);
    tdm_load_2d(ldsbase,
                ((unsigned long long)gahi << 32) | galo,
                /*td0=*/U_, /*td1=*/xl, /*tile0=*/U_, /*tile1=*/L_,
                /*stride0=*/U_,
                /*flags: 4B | pad_en | interval=7 | amount=1*/
                (2u << 16) | (1u << 20) | (7u << 22) | (1u << 25));
  }
  __syncthreads();  // qsh visible (window DMA still in flight)

  // ---- q2 = query @ W2 + b2 (tiny), overlapped with window DMA ----
  {
    float a0 = b2[tid], a1 = b2[tid + 256];
    for (int k = 0; k < U_; ++k) {
      const float qk = qsh[k];
      a0 = fmaf(qk, W2[(size_t)k * U_ + tid],       a0);
      a1 = fmaf(qk, W2[(size_t)k * U_ + tid + 256], a1);
    }
    q2[tid]       = a0;
    q2[tid + 256] = a1;
  }

  // ---- WMMA GEMM over 32 N-panels of W1, fused tanh+V reduction ----
  float sp[8];
#pragma unroll
  for (int r = 0; r < 8; ++r) sp[r] = 0.f;

  const float* arow0 = att + mrow * ATT_STRIDE + 2 * kh;  // u in [0,256)
  const float* arow1 = arow0 + 2;                         // u in [256,512): +2 pad
  const float* bb    = panel + lmod + 32 * kh;            // panel[k][n], stride 16

  for (int j = 0; j < 32; ++j) {
    const int n0 = 16 * j;
    // TDM: stage W1 panel [512 x 16] at column n0 (no transpose, no pad)
    if (wave == 0) {
      unsigned long long ga = (unsigned long long)(uintptr_t)W1 +
                              (unsigned long long)(n0 * 4);
      tdm_load_2d(ldsbase + ATT_FLOATS * 4, ga,
                  /*td0=*/U_, /*td1=*/U_, /*tile0=*/16, /*tile1=*/U_,
                  /*stride0=*/U_, /*flags: 4B elems*/ (2u << 16));
      __builtin_amdgcn_s_wait_tensorcnt(0);  // covers window DMA too (j==0)
    }
    __syncthreads();  // panel (+ att, q2 on j==0) visible to all waves

    // 4 independent accumulator chains to hide XDL latency.
    v8f acc0 = {}, acc1 = {}, acc2 = {}, acc3 = {};
    for (int k0 = 0; k0 < 256; k0 += 16) {
      v2f a0 = *(const v2f*)(arow0 + k0 + 0);
      v2f a1 = *(const v2f*)(arow0 + k0 + 4);
      v2f a2 = *(const v2f*)(arow0 + k0 + 8);
      v2f a3 = *(const v2f*)(arow0 + k0 + 12);
      v2f b0v = {bb[(k0 + 0) * 16],  bb[(k0 + 1) * 16]};
      v2f b1v = {bb[(k0 + 4) * 16],  bb[(k0 + 5) * 16]};
      v2f b2v = {bb[(k0 + 8) * 16],  bb[(k0 + 9) * 16]};
      v2f b3v = {bb[(k0 + 12) * 16], bb[(k0 + 13) * 16]};
      acc0 = __builtin_amdgcn_wmma_f32_16x16x4_f32(false, a0, false, b0v, (short)0, acc0, false, false);
      acc1 = __builtin_amdgcn_wmma_f32_16x16x4_f32(false, a1, false, b1v, (short)0, acc1, false, false);
      acc2 = __builtin_amdgcn_wmma_f32_16x16x4_f32(false, a2, false, b2v, (short)0, acc2, false, false);
      acc3 = __builtin_amdgcn_wmma_f32_16x16x4_f32(false, a3, false, b3v, (short)0, acc3, false, false);
    }
    for (int k0 = 256; k0 < 512; k0 += 16) {
      v2f a0 = *(const v2f*)(arow1 + k0 + 0);
      v2f a1 = *(const v2f*)(arow1 + k0 + 4);
      v2f a2 = *(const v2f*)(arow1 + k0 + 8);
      v2f a3 = *(const v2f*)(arow1 + k0 + 12);
      v2f b0v = {bb[(k0 + 0) * 16],  bb[(k0 + 1) * 16]};
      v2f b1v = {bb[(k0 + 4) * 16],  bb[(k0 + 5) * 16]};
      v2f b2v = {bb[(k0 + 8) * 16],  bb[(k0 + 9) * 16]};
      v2f b3v = {bb[(k0 + 12) * 16], bb[(k0 + 13) * 16]};
      acc0 = __builtin_amdgcn_wmma_f32_16x16x4_f32(false, a0, false, b0v, (short)0, acc0, false, false);
      acc1 = __builtin_amdgcn_wmma_f32_16x16x4_f32(false, a1, false, b1v, (short)0, acc1, false, false);
      acc2 = __builtin_amdgcn_wmma_f32_16x16x4_f32(false, a2, false, b2v, (short)0, acc2, false, false);
      acc3 = __builtin_amdgcn_wmma_f32_16x16x4_f32(false, a3, false, b3v, (short)0, acc3, false, false);
    }
    const v8f acc = (acc0 + acc1) + (acc2 + acc3);

    // Fused epilogue: sp[r] += tanh(C + b1[n] + q2[n]) * V[n]
    const int n     = n0 + lmod;
    const float bia = b1[n] + q2[n];
    const float vn  = V[n];
#pragma unroll
    for (int r = 0; r < 8; ++r)
      sp[r] += tanhf(acc[r] + bia) * vn;

    __syncthreads();  // before next panel DMA overwrites the buffer
  }

  // ---- reduce score partials (C layout: VGPR r -> rows 16w+r / 16w+8+r) ----
#pragma unroll
  for (int r = 0; r < 8; ++r) {
    float t = sp[r];
    t += __shfl_xor(t, 1);
    t += __shfl_xor(t, 2);
    t += __shfl_xor(t, 4);
    t += __shfl_xor(t, 8);
    if (lane == 0)  score[16 * wave + r]     = t;
    if (lane == 16) score[16 * wave + 8 + r] = t;
  }
  __syncthreads();

  // ---- softmax over L=128 (bV cancels in softmax) ----
  if (tid == 0) {
    float mx = score[0];
    for (int l = 1; l < L_; ++l) mx = fmaxf(mx, score[l]);
    float s = 0.f;
    for (int l = 0; l < L_; ++l) s += expf(score[l] - mx);
    red[0] = mx;
    red[1] = 1.0f / s;
  }
  __syncthreads();
  if (tid < L_) {
    const float w = expf(score[tid] - red[0]) * red[1];
    wts[tid] = w;
    out[B_ * U_ + b * L_ + tid] = w;   // attention_weights output
  }
  __syncthreads();

  // ---- context = sum_l w[l] * att[l,:]  (note +258 = 256 + TDM pad) ----
  float c0 = 0.f, c1 = 0.f;
  for (int l = 0; l < L_; ++l) {
    const float w = wts[l];
    c0 = fmaf(w, att[l * ATT_STRIDE + tid],       c0);
    c1 = fmaf(w, att[l * ATT_STRIDE + tid + 258], c1);
  }
  out[b * U_ + tid]       = c0;
  out[b * U_ + tid + 256] = c1;
}

extern "C" void kernel_launch(void* const* d_in, const int* in_sizes, int n_in,
                              void* d_out, int out_size, void* d_ws, size_t ws_size,
                              hipStream_t stream) {
  const float* query  = (const float*)d_in[0];
  const float* values = (const float*)d_in[1];
  const int*   pos    = (const int*)d_in[2];
  const float* W1     = (const float*)d_in[3];
  const float* b1     = (const float*)d_in[4];
  const float* W2     = (const float*)d_in[5];
  const float* b2     = (const float*)d_in[6];
  const float* V      = (const float*)d_in[7];
  // d_in[8] = bV: cancels in softmax, unused
  float* out = (float*)d_out;

  const size_t shmem =
      (size_t)(ATT_FLOATS + PAN_FLOATS + U_ + U_ + L_ + L_ + 2) * sizeof(float);

  bahdanau_wmma_kernel<<<dim3(B_), dim3(256), shmem, stream>>>(
      query, values, pos, W1, b1, W2, b2, V, out);
}